// LuongAttnDecoderRNN_90623809945975
// MI455X (gfx1250) — compile-verified
//
#include <hip/hip_runtime.h>
#include <hip/hip_bf16.h>
#include <math.h>

typedef float v2f __attribute__((ext_vector_type(2)));
typedef float v8f __attribute__((ext_vector_type(8)));

#define Bsz 64
#define Hdim 1024
#define Vocab 50257
#define Lenc 1024
#define NLay 2

#define KC 32
#define KCP 36   // padded LDS row stride (floats): 144B -> 16B aligned, 16 distinct banks

// Async global->LDS 128-bit copy (ASYNCcnt-tracked). lds = LDS byte offset,
// g = per-lane global address. Inline asm per CDNA5_HIP.md (builtin arity is
// toolchain-dependent; asm is portable).
__device__ __forceinline__ void async_ld128(unsigned lds, const void* g) {
    asm volatile("global_load_async_to_lds_b128 %0, %1, off"
                 :: "v"(lds), "v"((unsigned long long)(uintptr_t)g)
                 : "memory");
}

// ---------------------------------------------------------------------------
// Generic fp32 WMMA GEMM:  D[m,n] = epi( sum_k A[m,k] * W[n,k] (+ bias[n]) )
// A: 64 x K (row stride lda), W: Ntot x K row-major, D: 64 x Ntot (row stride ldd)
// mode 0: raw   mode 1: tanh(x+bias)   mode 2: x+bias
// Block: 256 thr (8 waves) -> 64x32 tile; grid.x = ceil(Ntot/32)
// Double-buffered LDS staging via global_load_async_to_lds_b128.
// ---------------------------------------------------------------------------
__global__ __launch_bounds__(256) void gemm_wmma_f32(
    const float* __restrict__ A, int lda,
    const float* __restrict__ W,
    const float* __restrict__ bias,
    float* __restrict__ D, int ldd,
    int Ntot, int K, int mode)
{
    __shared__ __attribute__((aligned(16))) float As[2][64 * KCP];
    __shared__ __attribute__((aligned(16))) float Ws[2][32 * KCP];

    const int tid  = threadIdx.x;
    const int lane = tid & 31;
    const int wv   = tid >> 5;              // wave 0..7
    const int m0   = (wv & 3) << 4;         // 0,16,32,48
    const int n0   = (wv >> 2) << 4;        // 0,16
    const int nblk = blockIdx.x << 5;       // 32 columns per block
    const int hi   = lane >> 4;             // 0 | 1  (lane half)
    const int lm   = lane & 15;

    // cooperative-load indices
    const int am = tid >> 2;                // 0..63 (A row)
    const int ak = (tid & 3) << 3;          // 0,8,16,24
    const int wn = tid >> 3;                // 0..31 (W row in tile)
    const int wk = (tid & 7) << 2;          // 0..28 step 4
    int wrow = nblk + wn; if (wrow >= Ntot) wrow = Ntot - 1;

    const float* Arow = A + (size_t)am * lda;      // this thread's A source row
    const float* Wrow = W + (size_t)wrow * K;      // this thread's W source row

    // per-thread LDS byte offsets (flat-address low 32 bits == LDS offset)
    const unsigned ldsA0 = (unsigned)(uintptr_t)&As[0][am * KCP + ak];
    const unsigned ldsA1 = (unsigned)(uintptr_t)&As[1][am * KCP + ak];
    const unsigned ldsW0 = (unsigned)(uintptr_t)&Ws[0][wn * KCP + wk];
    const unsigned ldsW1 = (unsigned)(uintptr_t)&Ws[1][wn * KCP + wk];

    v8f acc = {};

    const int nchunks = K / KC;

    // prologue: issue chunk 0 into buffer 0
    async_ld128(ldsA0,      Arow + ak);
    async_ld128(ldsA0 + 16, Arow + ak + 4);
    async_ld128(ldsW0,      Wrow + wk);

    for (int it = 0; it < nchunks; ++it) {
        const int buf = it & 1;
        if (it + 1 < nchunks) {
            const int kn = (it + 1) * KC;
            const unsigned la = buf ? ldsA0 : ldsA1;
            const unsigned lw = buf ? ldsW0 : ldsW1;
            async_ld128(la,      Arow + kn + ak);
            async_ld128(la + 16, Arow + kn + ak + 4);
            async_ld128(lw,      Wrow + kn + wk);
            // 3 newest (next chunk) may remain outstanding; current chunk done
            asm volatile("s_wait_asynccnt 0x3" ::: "memory");
        } else {
            asm volatile("s_wait_asynccnt 0x0" ::: "memory");
        }
        __syncthreads();   // all waves' current-chunk data visible in LDS

        const float* Ab = &As[buf][0];
        const float* Wb = &Ws[buf][0];
        #pragma unroll
        for (int kk = 0; kk < KC; kk += 4) {
            // A frag: lane -> A[m0+lm][kk+2*hi .. +1]
            v2f a = *(const v2f*)(&Ab[(m0 + lm) * KCP + kk + 2 * hi]);
            // B frag: lane -> W[n0+lm][kk+2*hi .. +1]  (B[k][n] = W[n][k])
            v2f b = *(const v2f*)(&Wb[(n0 + lm) * KCP + kk + 2 * hi]);
            acc = __builtin_amdgcn_wmma_f32_16x16x4_f32(
                false, a, false, b, (short)0, acc, false, false);
        }
        __syncthreads();   // safe to overwrite the other buffer next iteration
    }

    const int n = nblk + n0 + lm;
    if (n < Ntot) {
        float bv = (mode != 0 && bias) ? bias[n] : 0.0f;
        #pragma unroll
        for (int r = 0; r < 8; ++r) {
            int m = m0 + r + (hi << 3);     // C/D layout: VGPR r -> M=r (lo) / r+8 (hi)
            float v = acc[r];
            if (mode == 1)      v = tanhf(v + bv);
            else if (mode == 2) v = v + bv;
            D[(size_t)m * ldd + n] = v;
        }
    }
}

// ---------------------------------------------------------------------------
// Embedding gather: x[b,:] = emb[idx[b],:]   (grid = B, block = 256, float4)
// ---------------------------------------------------------------------------
__global__ __launch_bounds__(256) void embed_gather(
    const int* __restrict__ idx, const float* __restrict__ emb,
    float* __restrict__ x)
{
    const int b = blockIdx.x;
    const float4* src = (const float4*)(emb + (size_t)idx[b] * Hdim);
    ((float4*)(x + (size_t)b * Hdim))[threadIdx.x] = src[threadIdx.x];
}

// ---------------------------------------------------------------------------
// GRU gate fusion (PyTorch order r,z,n). One thread per (b, j).
// ---------------------------------------------------------------------------
__global__ __launch_bounds__(256) void gru_gate(
    const float* __restrict__ gx, const float* __restrict__ gh,
    const float* __restrict__ bih, const float* __restrict__ bhh,
    const float* __restrict__ hprev,
    float* __restrict__ out1,                  // new hidden (ld = Hdim)
    float* __restrict__ out2, int ld2)         // second copy (next-layer x / concat)
{
    const int idx = blockIdx.x * 256 + threadIdx.x;
    const int b = idx >> 10;
    const int j = idx & (Hdim - 1);
    const float* gxr = gx + (size_t)b * 3 * Hdim;
    const float* ghr = gh + (size_t)b * 3 * Hdim;

    float xr = gxr[j]            + bih[j];
    float xz = gxr[Hdim + j]     + bih[Hdim + j];
    float xn = gxr[2 * Hdim + j] + bih[2 * Hdim + j];
    float hr = ghr[j]            + bhh[j];
    float hz = ghr[Hdim + j]     + bhh[Hdim + j];
    float hn = ghr[2 * Hdim + j] + bhh[2 * Hdim + j];

    float r = 1.0f / (1.0f + __expf(-(xr + hr)));
    float z = 1.0f / (1.0f + __expf(-(xz + hz)));
    float n = tanhf(xn + r * hn);
    float hp = hprev[(size_t)b * Hdim + j];
    float h = (1.0f - z) * n + z * hp;

    out1[(size_t)b * Hdim + j] = h;
    out2[(size_t)b * ld2 + j]  = h;
}

// ---------------------------------------------------------------------------
// Fused dot-attention per batch row: energies -> softmax -> context.
// Grid = B, block = 256 (8 waves). Streams enc[:,b,:] (4 MB) once from HBM;
// the context pass re-reads it from L2.
// ---------------------------------------------------------------------------
__global__ __launch_bounds__(256) void attention_fused(
    const float* __restrict__ q,        // rnn_out, (B, H)
    const float* __restrict__ enc,      // (L, B, H)
    float* __restrict__ attn_out,       // (B, L)
    float* __restrict__ ctx_out, int ldc)
{
    __shared__ __attribute__((aligned(16))) float q_s[Hdim];
    __shared__ float e_s[Lenc];
    __shared__ float red[256];

    const int b    = blockIdx.x;
    const int tid  = threadIdx.x;
    const int lane = tid & 31;
    const int wv   = tid >> 5;

    // load query row into LDS
    ((float4*)q_s)[tid] = ((const float4*)(q + (size_t)b * Hdim))[tid];
    __syncthreads();

    // Phase 1: e[l] = q . enc[l,b,:]   (one l per wave per step)
    for (int l = wv; l < Lenc; l += 8) {
        const float4* r4 = (const float4*)(enc + ((size_t)l * Bsz + b) * Hdim);
        const float4* q4 = (const float4*)q_s;
        float p = 0.0f;
        #pragma unroll
        for (int t = 0; t < 8; ++t) {
            float4 ev = r4[lane + 32 * t];
            float4 qv = q4[lane + 32 * t];
            p += ev.x * qv.x + ev.y * qv.y + ev.z * qv.z + ev.w * qv.w;
        }
        #pragma unroll
        for (int o = 16; o > 0; o >>= 1) p += __shfl_down(p, o, 32);
        if (lane == 0) e_s[l] = p;
    }
    __syncthreads();

    // Phase 2: softmax over L=1024
    float lm = -3.4e38f;
    #pragma unroll
    for (int i = 0; i < 4; ++i) lm = fmaxf(lm, e_s[tid + 256 * i]);
    red[tid] = lm;  __syncthreads();
    for (int s = 128; s > 0; s >>= 1) {
        if (tid < s) red[tid] = fmaxf(red[tid], red[tid + s]);
        __syncthreads();
    }
    const float mx = red[0];  __syncthreads();

    float ls = 0.0f;
    #pragma unroll
    for (int i = 0; i < 4; ++i) ls += __expf(e_s[tid + 256 * i] - mx);
    red[tid] = ls;  __syncthreads();
    for (int s = 128; s > 0; s >>= 1) {
        if (tid < s) red[tid] += red[tid + s];
        __syncthreads();
    }
    const float inv_s = 1.0f / red[0];  __syncthreads();

    #pragma unroll
    for (int i = 0; i < 4; ++i) {
        int l = tid + 256 * i;
        float w = __expf(e_s[l] - mx) * inv_s;
        attn_out[(size_t)b * Lenc + l] = w;
        e_s[l] = w;
    }
    __syncthreads();

    // Phase 3: context[b, 4t..4t+3] = sum_l w[l] * enc[l, b, 4t..4t+3]
    float4 c = make_float4(0.f, 0.f, 0.f, 0.f);
    const float4* eb4 = (const float4*)enc;
    const size_t rowq = (size_t)Hdim / 4;
    for (int l = 0; l < Lenc; ++l) {
        float w = e_s[l];
        float4 ev = eb4[((size_t)l * Bsz + b) * rowq + tid];
        c.x += w * ev.x; c.y += w * ev.y; c.z += w * ev.z; c.w += w * ev.w;
    }
    *(float4*)(ctx_out + (size_t)b * ldc + 4 * tid) = c;
}

// ---------------------------------------------------------------------------
extern "C" void kernel_launch(void* const* d_in, const int* in_sizes, int n_in,
                              void* d_out, int out_size, void* d_ws, size_t ws_size,
                              hipStream_t stream) {
    const int*   input_seq   = (const int*)  d_in[0];
    const float* last_hidden = (const float*)d_in[1];
    const float* enc         = (const float*)d_in[2];
    const float* emb         = (const float*)d_in[3];
    const float* W_ih        = (const float*)d_in[4];
    const float* W_hh        = (const float*)d_in[5];
    const float* b_ih        = (const float*)d_in[6];
    const float* b_hh        = (const float*)d_in[7];
    const float* W_cat       = (const float*)d_in[8];
    const float* b_cat       = (const float*)d_in[9];
    const float* W_out       = (const float*)d_in[10];
    const float* b_out       = (const float*)d_in[11];

    float* out_logits = (float*)d_out;                              // (B, V)
    float* out_hidden = out_logits + (size_t)Bsz * Vocab;           // (NL, B, H)
    float* out_attn   = out_hidden + (size_t)NLay * Bsz * Hdim;     // (B, 1, L)

    float* ws   = (float*)d_ws;
    float* x0   = ws;                          // (B, H)   layer input
    float* gx   = x0  + (size_t)Bsz * Hdim;    // (B, 3H)
    float* gh   = gx  + (size_t)Bsz * 3 * Hdim;// (B, 3H)
    float* cat  = gh  + (size_t)Bsz * 3 * Hdim;// (B, 2H)  [rnn_out | context]
    float* cout = cat + (size_t)Bsz * 2 * Hdim;// (B, H)

    embed_gather<<<Bsz, 256, 0, stream>>>(input_seq, emb, x0);

    for (int l = 0; l < NLay; ++l) {
        const float* hprev = last_hidden + (size_t)l * Bsz * Hdim;
        const float* Wi = W_ih + (size_t)l * 3 * Hdim * Hdim;
        const float* Wh = W_hh + (size_t)l * 3 * Hdim * Hdim;
        gemm_wmma_f32<<<(3 * Hdim) / 32, 256, 0, stream>>>(
            x0, Hdim, Wi, nullptr, gx, 3 * Hdim, 3 * Hdim, Hdim, 0);
        gemm_wmma_f32<<<(3 * Hdim) / 32, 256, 0, stream>>>(
            hprev, Hdim, Wh, nullptr, gh, 3 * Hdim, 3 * Hdim, Hdim, 0);
        float* out2 = (l == 0) ? x0 : cat;     // next-layer x, or concat[:, :H]
        int    ld2  = (l == 0) ? Hdim : 2 * Hdim;
        gru_gate<<<(Bsz * Hdim) / 256, 256, 0, stream>>>(
            gx, gh, b_ih + (size_t)l * 3 * Hdim, b_hh + (size_t)l * 3 * Hdim,
            hprev, out_hidden + (size_t)l * Bsz * Hdim, out2, ld2);
    }

    attention_fused<<<Bsz, 256, 0, stream>>>(
        out_hidden + (size_t)(NLay - 1) * Bsz * Hdim, enc,
        out_attn, cat + Hdim, 2 * Hdim);

    gemm_wmma_f32<<<Hdim / 32, 256, 0, stream>>>(
        cat, 2 * Hdim, W_cat, b_cat, cout, Hdim, Hdim, 2 * Hdim, 1);

    gemm_wmma_f32<<<(Vocab + 31) / 32, 256, 0, stream>>>(
        cout, Hdim, W_out, b_out, out_logits, Vocab, Vocab, Hdim, 2);
}